// StructureEncoder1D_7009386627271
// MI455X (gfx1250) — compile-verified
//
#include <hip/hip_runtime.h>

// StructureEncoder1D on gfx1250 (MI455X, wave32).
//
// Per joint i: h = relu([x_i, parent_feat(6)] @ W1[i]^T + b1[i])   (7 -> 7)
//             f = relu(h @ W2[i]^T + b2[i])                        (7 -> 6)
// Output: out[b][i*6 + c], i = 0..23 (PARENTS[i] < i, so index order is topo order).
//
// WMMA mapping (V_WMMA_F32_16X16X4_F32, D = A(16x4) * B(4x16) + C):
//   A rows (M)  = output features; rows 0..6 = W1ext (bias as col 7),
//                 rows 8..13 = W2ext (bias as col 7), rows 7,14,15 = 0.
//   B cols (N)  = 16 samples; B rows (K) = input vector [x, pf0..pf5, 1.0].
//   K = 8  -> two chained WMMAs (k 0..3 and k 4..7).
//   Layer 1 result: D rows 0..6  -> VGPR c, lanes 0..15.
//   Layer 2 result: D rows 8..13 -> VGPR c, lanes 16..31 (same A registers!).
//
// ISA register layouts (CDNA5 ISA, 7.12.2):
//   A(m,k): lane = m + 16*(k>>1), vgpr = k&1
//   B(k,n): lane = n + 16*(k>>1), vgpr = k&1
//   D(m,n): vgpr = m%8, lane = (m/8)*16 + n
//
// Lane-half moves use DS_SWIZZLE_B32 SWAPX16. CRITICAL: ds_swizzle is a
// convergent cross-lane op and reads from inactive lanes return 0, so every
// swizzle below is executed unconditionally (full EXEC) and only the *select*
// of its result is lane-dependent (v_cndmask). Never place swapx16() inside
// a lane-divergent ternary arm / if-body.

#define NJ   24
#define FEAT 6
#define HID  7

typedef float v2f __attribute__((ext_vector_type(2)));
typedef float v8f __attribute__((ext_vector_type(8)));

__device__ __forceinline__ float swapx16(float v) {
  // lane <- lane ^ 16 (must run with all 32 lanes active)
  return __int_as_float(__builtin_amdgcn_ds_swizzle(__float_as_int(v), 0x401F));
}

__device__ __forceinline__ float relu1(float v) {
  // v_med3_f32(v, 0, +inf) == clamp to [0, inf): single-instruction ReLU
  return __builtin_amdgcn_fmed3f(v, 0.0f, __builtin_inff());
}

// One element of the combined (W1ext / W2ext) A matrix for joint i.
__device__ __forceinline__ float welem(const float* __restrict__ W1,
                                       const float* __restrict__ b1,
                                       const float* __restrict__ W2,
                                       const float* __restrict__ b2,
                                       int i, int m, int k) {
  if (m < HID) {                       // rows 0..6: layer-1 weights
    return (k < HID) ? W1[i * HID * HID + m * HID + k] : b1[i * HID + m];
  }
  if (m >= 8 && m < 8 + FEAT) {        // rows 8..13: layer-2 weights
    int r = m - 8;
    return (k < HID) ? W2[i * FEAT * HID + r * HID + k] : b2[i * FEAT + r];
  }
  return 0.0f;                         // rows 7, 14, 15: padding
}

__global__ __launch_bounds__(256) void se1d_wmma_kernel(
    const float* __restrict__ x,   // (B, 24)
    const float* __restrict__ W1,  // (24, 7, 7)
    const float* __restrict__ b1,  // (24, 7)
    const float* __restrict__ W2,  // (24, 6, 7)
    const float* __restrict__ b2,  // (24, 6)
    float* __restrict__ out,       // (B, 144)
    int Btot, int NT) {
  constexpr int kPar[NJ] = {-1, 0, 0, 0, 1, 2, 3, 4, 5, 6, 7, 8,
                            9, 9, 9, 12, 13, 14, 16, 17, 18, 19, 20, 21};

  const int lane = threadIdx.x & 31;
  const int m    = lane & 15;          // A-row index for this lane (also sample slot)
  const bool hi  = lane >= 16;
  const int kb   = hi ? 2 : 0;         // per-lane K base within a 16x16x4 block

  const int wavesPerBlock = blockDim.x >> 5;
  const int gwave = blockIdx.x * wavesPerBlock + (threadIdx.x >> 5);
  const int nwave = gridDim.x * wavesPerBlock;

  // ---- gather per-joint combined weight matrices into A layout (resident) ----
  float wa[NJ][4];
#pragma unroll
  for (int i = 0; i < NJ; ++i) {
    wa[i][0] = welem(W1, b1, W2, b2, i, m, kb + 0);  // WMMA #0, vgpr 0 (k=0|2)
    wa[i][1] = welem(W1, b1, W2, b2, i, m, kb + 1);  // WMMA #0, vgpr 1 (k=1|3)
    wa[i][2] = welem(W1, b1, W2, b2, i, m, kb + 4);  // WMMA #1, vgpr 0 (k=4|6)
    wa[i][3] = welem(W1, b1, W2, b2, i, m, kb + 5);  // WMMA #1, vgpr 1 (k=5|7)
  }

  // ---- grid-stride over 16-sample tiles ----
  for (int tile = gwave; tile < NT; tile += nwave) {
    const int sample = tile * 16 + m;
    const int sclamp = (sample < Btot) ? sample : (Btot - 1);
    const bool svalid = sample < Btot;

    // load this lane's x row (24 floats, 16B aligned)
    float xv[NJ];
    {
      const float4* xr = (const float4*)(x + (size_t)sclamp * NJ);
#pragma unroll
      for (int q = 0; q < NJ / 4; ++q) {
        float4 v = xr[q];
        xv[4 * q + 0] = v.x; xv[4 * q + 1] = v.y;
        xv[4 * q + 2] = v.z; xv[4 * q + 3] = v.w;
      }
    }

    const size_t outRow = (size_t)sample * (NJ * FEAT);
    float f[NJ][FEAT];  // joint features: valid in lanes 16..31 (vgpr c = feat c)

#pragma unroll
    for (int ji = 0; ji < NJ; ++ji) {
      const int pi = kPar[ji];  // compile-time constant after unroll

      // ---- build layer-1 B operand: rows k = [x, pf0..pf5, 1] ----
      float Bv0, Bv1, Bv2, Bv3;
      if (pi < 0) {
        Bv0 = hi ? 0.0f : xv[ji];      // k=0: x      | k=2: pf1 (=0)
        Bv1 = 0.0f;                    // k=1: pf0    | k=3: pf2
        Bv2 = 0.0f;                    // k=4: pf3    | k=6: pf5
        Bv3 = hi ? 1.0f : 0.0f;        // k=5: pf4    | k=7: 1.0 (bias)
      } else {
        // swizzles first, unconditionally, at full EXEC:
        float sw0 = swapx16(f[pi][0]); // hi-lane pf0 -> lo lanes
        float sw3 = swapx16(f[pi][3]); // hi-lane pf3 -> lo lanes
        float sw4 = swapx16(f[pi][4]); // hi-lane pf4 -> lo lanes
        Bv0 = hi ? f[pi][1] : xv[ji];
        Bv1 = hi ? f[pi][2] : sw0;
        Bv2 = hi ? f[pi][5] : sw3;
        Bv3 = hi ? 1.0f     : sw4;
      }

      v2f a0 = {wa[ji][0], wa[ji][1]};
      v2f a1 = {wa[ji][2], wa[ji][3]};
      v2f bl0 = {Bv0, Bv1};
      v2f bl1 = {Bv2, Bv3};

      // ---- layer 1: D rows 0..6 (lo lanes) = pre-activation h ----
      v8f d = {0.f, 0.f, 0.f, 0.f, 0.f, 0.f, 0.f, 0.f};
      d = __builtin_amdgcn_wmma_f32_16x16x4_f32(false, a0, false, bl0,
                                                (short)0, d, false, false);
      d = __builtin_amdgcn_wmma_f32_16x16x4_f32(false, a1, false, bl1,
                                                (short)0, d, false, false);
      float h0 = relu1(d[0]), h1 = relu1(d[1]), h2 = relu1(d[2]);
      float h3 = relu1(d[3]), h4 = relu1(d[4]), h5 = relu1(d[5]);
      float h6 = relu1(d[6]);

      // ---- layer-2 B operand: rows k = [h0..h6, 1] (h lives in lo lanes) ----
      // swizzles unconditionally at full EXEC, then select:
      float t2 = swapx16(h2);          // lo-lane h2 -> hi lanes
      float t3 = swapx16(h3);
      float t6 = swapx16(h6);
      v2f bh0 = {hi ? t2 : h0, hi ? t3 : h1};
      v2f bh1 = {hi ? t6 : h4, hi ? 1.0f : h5};

      // ---- layer 2: same A registers; D rows 8..13 (hi lanes) = features ----
      v8f d2 = {0.f, 0.f, 0.f, 0.f, 0.f, 0.f, 0.f, 0.f};
      d2 = __builtin_amdgcn_wmma_f32_16x16x4_f32(false, a0, false, bh0,
                                                 (short)0, d2, false, false);
      d2 = __builtin_amdgcn_wmma_f32_16x16x4_f32(false, a1, false, bh1,
                                                 (short)0, d2, false, false);
#pragma unroll
      for (int c = 0; c < FEAT; ++c) f[ji][c] = relu1(d2[c]);

      // ---- store this joint's 6 features (valid in hi lanes only) ----
      if (hi && svalid) {
        float* p = out + outRow + ji * FEAT;  // dword offset even -> 8B aligned
        float2* p2 = (float2*)p;
        p2[0] = float2{f[ji][0], f[ji][1]};
        p2[1] = float2{f[ji][2], f[ji][3]};
        p2[2] = float2{f[ji][4], f[ji][5]};
      }
    }
  }
}

extern "C" void kernel_launch(void* const* d_in, const int* in_sizes, int n_in,
                              void* d_out, int out_size, void* d_ws, size_t ws_size,
                              hipStream_t stream) {
  (void)n_in; (void)d_ws; (void)ws_size; (void)out_size;
  const float* x  = (const float*)d_in[0];
  const float* W1 = (const float*)d_in[1];
  const float* b1 = (const float*)d_in[2];
  const float* W2 = (const float*)d_in[3];
  const float* b2 = (const float*)d_in[4];
  float* out = (float*)d_out;

  const int Btot = in_sizes[0] / NJ;          // 500000
  const int NT   = (Btot + 15) / 16;          // 16-sample tiles
  const int wavesPerBlock = 256 / 32;
  int blocks = (NT + wavesPerBlock - 1) / wavesPerBlock;
  if (blocks > 768) blocks = 768;             // grid-stride; keep waves resident so
                                              // the 96-VGPR weight cache amortizes
  se1d_wmma_kernel<<<blocks, 256, 0, stream>>>(x, W1, b1, W2, b2, out, Btot, NT);
}